// ConcordanceCC_47562467836084
// MI455X (gfx1250) — compile-verified
//
#include <hip/hip_runtime.h>
#include <stdint.h>

// ---------------------------------------------------------------------------
// ConcordanceCC on MI455X (gfx1250): single-pass masked moment reduction.
//   ws layout: per segment s in [0,1024): ws[s*8 + k], k = {S1,S2,L,Stt,Spp,Stp}
// ---------------------------------------------------------------------------

#define T_LEN   65536
#define B_ROWS  256
#define SEGS    4
#define SEG_LEN (T_LEN / SEGS)            // 16384 elements per block
#define THREADS 256                        // 8 wave32s per block
#define ITERS   (SEG_LEN / (THREADS * 4))  // 16 x float4 per thread

// ---- gfx1250 async global->LDS path (probe-confirmed builtin) -------------
#if defined(__has_builtin)
#  if __has_builtin(__builtin_amdgcn_global_load_async_to_lds_b128)
#    define CCC_ASYNC_LDS 1
#  endif
#endif

#if defined(CCC_ASYNC_LDS)
// Builtin prototype (from clang diagnostic): param0 = AS1 (global) int4*,
// param1 = AS3 (LDS) int4*, then two immediate ints (offset, cpol).
typedef int ccc_v4i __attribute__((vector_size(16)));
typedef __attribute__((address_space(1))) ccc_v4i* ccc_gas_ptr;
typedef __attribute__((address_space(3))) ccc_v4i* ccc_las_ptr;
// AS1 value == flat address; AS3 value == low 32 bits of the flat address
// (addrspacecast generic->LDS is a truncate on amdgcn).
#define CCC_GP(p) ((ccc_gas_ptr)(uintptr_t)(p))
#define CCC_LP(p) ((ccc_las_ptr)(uint32_t)(uintptr_t)(p))
#endif

__device__ __forceinline__ void ccc_acc6(float t, float p, float m,
                                         float& s1, float& s2, float& sl,
                                         float& stt, float& spp, float& stp)
{
    const float mt = t * m;                // m is 0/1 so m*m == m
    const float mp = p * m;
    s1 += mt;
    s2 += mp;
    sl += m;
    stt = fmaf(mt, t, stt);                // t*t*m
    spp = fmaf(mp, p, spp);                // p*p*m
    stp = fmaf(mt, p, stp);                // t*p*m
}

__global__ __launch_bounds__(THREADS)
void ccc_partial_kernel(const float* __restrict__ yt,
                        const float* __restrict__ yp,
                        const int*   __restrict__ mk,
                        float*       __restrict__ ws)
{
    const int seg  = blockIdx.x;           // 0 .. B_ROWS*SEGS-1  (row = seg/SEGS)
    const int tid  = threadIdx.x;
    const int base = seg * SEG_LEN;

    float s1 = 0.f, s2 = 0.f, sl = 0.f, stt = 0.f, spp = 0.f, stp = 0.f;

#if defined(CCC_ASYNC_LDS)
    // Double-buffered async staging: 2 bufs x 3 streams x 256 lanes x 16B = 24 KB LDS.
    __shared__ float4 sb[2][3][THREADS];
    const float* gt = yt + base;
    const float* gp = yp + base;
    const float* gm = (const float*)mk + base;   // int32 mask: stage raw bits

    auto issue = [&](int buf, int it) {
        const int off = (it * THREADS + tid) * 4;        // element offset
        __builtin_amdgcn_global_load_async_to_lds_b128(CCC_GP(gt + off),
                                                       CCC_LP(&sb[buf][0][tid]), 0, 0);
        __builtin_amdgcn_global_load_async_to_lds_b128(CCC_GP(gp + off),
                                                       CCC_LP(&sb[buf][1][tid]), 0, 0);
        __builtin_amdgcn_global_load_async_to_lds_b128(CCC_GP(gm + off),
                                                       CCC_LP(&sb[buf][2][tid]), 0, 0);
    };

    issue(0, 0);
    for (int it = 0; it < ITERS; ++it) {
        const int cur = it & 1;
        if (it + 1 < ITERS) {
            // WAR guard: previous DS reads of buffer (cur^1) must land in VGPRs
            // before the async unit is allowed to overwrite it.
            asm volatile("s_wait_dscnt 0" ::: "memory");
            issue(cur ^ 1, it + 1);
            // Keep the 3 newest async loads in flight; current tile is complete.
            asm volatile("s_wait_asynccnt 3" ::: "memory");
        } else {
            asm volatile("s_wait_asynccnt 0" ::: "memory");
        }
        const float4 tv = sb[cur][0][tid];
        const float4 pv = sb[cur][1][tid];
        const float4 mr = sb[cur][2][tid];
        ccc_acc6(tv.x, pv.x, (float)__float_as_int(mr.x), s1, s2, sl, stt, spp, stp);
        ccc_acc6(tv.y, pv.y, (float)__float_as_int(mr.y), s1, s2, sl, stt, spp, stp);
        ccc_acc6(tv.z, pv.z, (float)__float_as_int(mr.z), s1, s2, sl, stt, spp, stp);
        ccc_acc6(tv.w, pv.w, (float)__float_as_int(mr.w), s1, s2, sl, stt, spp, stp);
    }
#else
    // Fallback: direct b128 streaming loads with software prefetch (global_prefetch_b8).
    const float4* t4 = (const float4*)(yt + base);
    const float4* p4 = (const float4*)(yp + base);
    const int4*   m4 = (const int4*)(mk + base);
    for (int it = 0; it < ITERS; ++it) {
        const int idx = it * THREADS + tid;
        if (it + 4 < ITERS) {
            __builtin_prefetch(t4 + idx + 4 * THREADS, 0, 0);
            __builtin_prefetch(p4 + idx + 4 * THREADS, 0, 0);
            __builtin_prefetch(m4 + idx + 4 * THREADS, 0, 0);
        }
        const float4 tv = t4[idx];
        const float4 pv = p4[idx];
        const int4   mv = m4[idx];
        ccc_acc6(tv.x, pv.x, (float)mv.x, s1, s2, sl, stt, spp, stp);
        ccc_acc6(tv.y, pv.y, (float)mv.y, s1, s2, sl, stt, spp, stp);
        ccc_acc6(tv.z, pv.z, (float)mv.z, s1, s2, sl, stt, spp, stp);
        ccc_acc6(tv.w, pv.w, (float)mv.w, s1, s2, sl, stt, spp, stp);
    }
#endif

    // ---- wave32 shuffle reduction, then cross-wave via LDS ----
#pragma unroll
    for (int o = 16; o > 0; o >>= 1) {
        s1  += __shfl_down(s1,  o, 32);
        s2  += __shfl_down(s2,  o, 32);
        sl  += __shfl_down(sl,  o, 32);
        stt += __shfl_down(stt, o, 32);
        spp += __shfl_down(spp, o, 32);
        stp += __shfl_down(stp, o, 32);
    }
    __shared__ float red[THREADS / 32][6];
    const int wave = tid >> 5;
    const int lane = tid & 31;
    if (lane == 0) {
        red[wave][0] = s1;  red[wave][1] = s2;  red[wave][2] = sl;
        red[wave][3] = stt; red[wave][4] = spp; red[wave][5] = stp;
    }
    __syncthreads();
    if (tid < 6) {
        float v = 0.f;
#pragma unroll
        for (int w = 0; w < THREADS / 32; ++w) v += red[w][tid];
        ws[seg * 8 + tid] = v;
    }
}

__global__ __launch_bounds__(B_ROWS)
void ccc_final_kernel(const float* __restrict__ ws, float* __restrict__ out)
{
    const int r = threadIdx.x;             // one thread per row
    double S1 = 0.0, S2 = 0.0, L = 0.0, Stt = 0.0, Spp = 0.0, Stp = 0.0;
#pragma unroll
    for (int s = 0; s < SEGS; ++s) {
        const float* p = ws + (size_t)(r * SEGS + s) * 8;
        S1 += p[0]; S2 += p[1]; L += p[2]; Stt += p[3]; Spp += p[4]; Stp += p[5];
    }
    const double mean_t = S1 / L;
    const double mean_p = S2 / L;
    const double denom  = L - 1.0;
    const double var_t  = (Stt - S1 * S1 / L) / denom;
    const double var_p  = (Spp - S2 * S2 / L) / denom;
    const double cov    = (Stp - S1 * S2 / L) / denom;
    // Faithful to reference: (mean_t - mean_p) * 2, NOT squared.
    double ccc = 2.0 * cov / (var_t + var_p + (mean_t - mean_p) * 2.0);

#pragma unroll
    for (int o = 16; o > 0; o >>= 1)
        ccc += __shfl_down(ccc, o, 32);
    __shared__ double rr[B_ROWS / 32];
    if ((r & 31) == 0) rr[r >> 5] = ccc;
    __syncthreads();
    if (r == 0) {
        double tot = 0.0;
#pragma unroll
        for (int w = 0; w < B_ROWS / 32; ++w) tot += rr[w];
        out[0] = (float)(tot / (double)B_ROWS);
    }
}

extern "C" void kernel_launch(void* const* d_in, const int* in_sizes, int n_in,
                              void* d_out, int out_size, void* d_ws, size_t ws_size,
                              hipStream_t stream)
{
    (void)in_sizes; (void)n_in; (void)out_size; (void)ws_size;
    const float* yt = (const float*)d_in[0];
    const float* yp = (const float*)d_in[1];
    const int*   mk = (const int*)d_in[2];
    float*       ws = (float*)d_ws;

    ccc_partial_kernel<<<B_ROWS * SEGS, THREADS, 0, stream>>>(yt, yp, mk, ws);
    ccc_final_kernel<<<1, B_ROWS, 0, stream>>>(ws, (float*)d_out);
}